// DecodePredictions_7249904796121
// MI455X (gfx1250) — compile-verified
//
#include <hip/hip_runtime.h>
#include <hip/hip_bf16.h>
#include <stdint.h>

#define BATCH   32
#define NANCH   8732
#define NCLS    90
#define NPRED   94          // 4 + 90
#define KSEL    200
#define KPAD    208         // 13 tiles of 16
#define NTILE   13
#define MAXDET  100
#define IOU_T   0.6f
#define SCORE_T 0.1f

typedef float v2f  __attribute__((ext_vector_type(2)));
typedef float v8f  __attribute__((ext_vector_type(8)));
typedef unsigned int uint32x4 __attribute__((ext_vector_type(4)));
typedef int   int32x4 __attribute__((ext_vector_type(4)));
typedef int   int32x8 __attribute__((ext_vector_type(8)));

// Monotonic order key: larger float -> larger uint
__device__ inline unsigned orderKey(float f) {
  unsigned u = __float_as_uint(f);
  return (u & 0x80000000u) ? ~u : (u | 0x80000000u);
}
__device__ inline float keyToFloat(unsigned k) {
  unsigned u = (k & 0x80000000u) ? (k & 0x7FFFFFFFu) : ~k;
  return __uint_as_float(u);
}

// ---------------------------------------------------------------- kernel 1
__global__ __launch_bounds__(256) void decode_boxes(
    const float* __restrict__ pred, const float* __restrict__ anchors,
    const float* __restrict__ var, float* __restrict__ boxes) {
  int idx = blockIdx.x * blockDim.x + threadIdx.x;
  if (idx >= BATCH * NANCH) return;
  int n = idx % NANCH;
  const float* p = pred + (size_t)idx * NPRED;
  float lx = p[0] * var[0], ly = p[1] * var[1];
  float lw = p[2] * var[2], lh = p[3] * var[3];
  float acx = anchors[n * 4 + 0], acy = anchors[n * 4 + 1];
  float aw  = anchors[n * 4 + 2], ah  = anchors[n * 4 + 3];
  float cx = lx * aw + acx, cy = ly * ah + acy;
  float w = __expf(lw) * aw, h = __expf(lh) * ah;
  float* o = boxes + (size_t)idx * 4;
  o[0] = cx - 0.5f * w; o[1] = cy - 0.5f * h;
  o[2] = cx + 0.5f * w; o[3] = cy + 0.5f * h;
}

// ---------------------------------------------------------------- kernel 2
// One workgroup per (b,c): exact top-K=200 on logits via 4-pass radix select
// (predictions are L2-resident: 105 MB < 192 MB global L2, so the strided
// multi-pass reads are cheap), then bitonic sort descending with low-index
// tie-break, then gather boxes + sigmoid.
__global__ __launch_bounds__(256) void topk_per_class(
    const float* __restrict__ pred, const float* __restrict__ boxes,
    float* __restrict__ candScore, float* __restrict__ candBox) {
  __shared__ unsigned hist[256];
  __shared__ unsigned sh_sel, sh_cum, sh_cntG, sh_cntE;
  __shared__ unsigned long long comp[256];
  int tid = threadIdx.x;
  int bc = blockIdx.x;
  int b = bc / NCLS, c = bc % NCLS;
  const float* base = pred + (size_t)b * NANCH * NPRED + 4 + c;

  unsigned prefix = 0, maskSo = 0, remK = KSEL;
  for (int shift = 24; shift >= 0; shift -= 8) {
    hist[tid] = 0; __syncthreads();
    for (int n = tid; n < NANCH; n += 256) {
      unsigned key = orderKey(base[(size_t)n * NPRED]);
      if ((key & maskSo) == prefix) atomicAdd(&hist[(key >> shift) & 255], 1u);
    }
    __syncthreads();
    if (tid == 0) {
      unsigned cum = 0; int sel = 0;
      for (int bin = 255; bin >= 0; bin--) {
        unsigned h = hist[bin];
        if (cum + h >= remK) { sel = bin; break; }
        cum += h;
      }
      sh_sel = (unsigned)sel; sh_cum = cum;
    }
    __syncthreads();
    prefix |= (sh_sel << shift); maskSo |= (0xFFu << shift); remK -= sh_cum;
    __syncthreads();
  }
  unsigned G = KSEL - remK;           // count strictly greater than pivot
  if (tid == 0) { sh_cntG = 0; sh_cntE = 0; }
  comp[tid] = 0ull;
  __syncthreads();
  for (int n = tid; n < NANCH; n += 256) {
    unsigned key = orderKey(base[(size_t)n * NPRED]);
    int slot = -1;
    if (key > prefix) slot = (int)atomicAdd(&sh_cntG, 1u);
    else if (key == prefix) {
      unsigned p = atomicAdd(&sh_cntE, 1u);
      if (p < remK) slot = (int)(G + p);
    }
    if (slot >= 0)
      comp[slot] = ((unsigned long long)key << 32) | (unsigned)(~(unsigned)n);
  }
  __syncthreads();
  // bitonic sort, descending (pad entries are 0 and sink to the end)
  for (int k = 2; k <= 256; k <<= 1)
    for (int j = k >> 1; j > 0; j >>= 1) {
      int ixj = tid ^ j;
      if (ixj > tid) {
        unsigned long long a = comp[tid], bb = comp[ixj];
        bool up = (tid & k) != 0;
        if (up ? (a > bb) : (a < bb)) { comp[tid] = bb; comp[ixj] = a; }
      }
      __syncthreads();
    }
  if (tid < KSEL) {
    unsigned long long e = comp[tid];
    unsigned key = (unsigned)(e >> 32);
    int n = (int)(~(unsigned)(e & 0xFFFFFFFFull));
    float logit = keyToFloat(key);
    float sc = 1.0f / (1.0f + __expf(-logit));
    size_t ob = (size_t)bc * KSEL + tid;
    candScore[ob] = sc;
    const float* bp = boxes + ((size_t)b * NANCH + n) * 4;
    float* op = candBox + ob * 4;
    op[0] = bp[0]; op[1] = bp[1]; op[2] = bp[2]; op[3] = bp[3];
  }
}

// ---------------------------------------------------------------- kernel 3
// One workgroup per (b,c). TDM stages the 200x4 candidate-box tile into LDS.
// Suppression bitmask built in 13x13 tiles of 16x16; the rank-2 part of the
// IoU union (area[i]+area[j]) is computed with V_WMMA_F32_16X16X4_F32:
//   D = A(16x4) x B(4x16),  A = [area_i, 1, 0, 0],  B = [1; area_j; 0; 0].
// Intersection (min/max) stays VALU. Greedy pass is then bit-parallel.
__global__ __launch_bounds__(256) void nms_per_class(
    float* __restrict__ candScore, const float* __restrict__ candBox) {
  __shared__ float stage[KSEL * 4];
  __shared__ float bx1[KPAD], by1[KPAD], bx2[KPAD], by2[KPAD];
  __shared__ float areaS[KPAD], scS[KPAD];
  __shared__ unsigned mask[KPAD][7];
  __shared__ unsigned keepW[7];
  int tid = threadIdx.x;
  int bc = blockIdx.x;
  const float* cbBase = candBox + (size_t)bc * KSEL * 4;

  if (tid == 0) {
#if __has_builtin(__builtin_amdgcn_tensor_load_to_lds)
    unsigned long long ga = (unsigned long long)(uintptr_t)cbBase;
    unsigned ldsOff = (unsigned)(uintptr_t)(&stage[0]);
    uint32x4 g0;
    g0.x = 1u;                                            // count=1, user D#
    g0.y = ldsOff;                                        // lds_addr
    g0.z = (unsigned)(ga & 0xFFFFFFFFull);                // global_addr[31:0]
    g0.w = (unsigned)((ga >> 32) & 0x01FFFFFFull) | (2u << 30); // addr[56:32] | type=2
    int32x8 g1;
    g1[0] = (int)(2u << 16);       // data_size=4B (bits 17:16)
    g1[1] = (int)(800u << 16);     // tensor_dim0[15:0] at bits 63:48
    g1[2] = (int)(1u << 16);       // tensor_dim0[31:16]=0, tensor_dim1[15:0]=1
    g1[3] = (int)(800u << 16);     // tile_dim0 at bits 127:112
    g1[4] = 1;                     // tile_dim1=1
    g1[5] = 800;                   // tensor_dim0_stride
    g1[6] = 0; g1[7] = 0;
    int32x4 gz4 = {0, 0, 0, 0};
#if defined(__clang_major__) && (__clang_major__ >= 23)
    int32x8 gz8 = {0, 0, 0, 0, 0, 0, 0, 0};
    __builtin_amdgcn_tensor_load_to_lds(g0, g1, gz4, gz4, gz8, 0);
#else
    __builtin_amdgcn_tensor_load_to_lds(g0, g1, gz4, gz4, 0);
#endif
#if __has_builtin(__builtin_amdgcn_s_wait_tensorcnt)
    __builtin_amdgcn_s_wait_tensorcnt(0);
#else
    asm volatile("s_wait_tensorcnt 0x0" ::: "memory");
#endif
#else
    for (int i = 0; i < KSEL * 4; i++) stage[i] = cbBase[i];
#endif
  }
  if (tid < KSEL)       scS[tid] = candScore[(size_t)bc * KSEL + tid];
  else if (tid < KPAD)  scS[tid] = 0.0f;
  __syncthreads();

  if (tid < KPAD) {
    float x1, y1, x2, y2;
    if (tid < KSEL) {
      x1 = stage[tid * 4 + 0]; y1 = stage[tid * 4 + 1];
      x2 = stage[tid * 4 + 2]; y2 = stage[tid * 4 + 3];
    } else { x1 = y1 = x2 = y2 = 0.0f; }
    bx1[tid] = x1; by1[tid] = y1; bx2[tid] = x2; by2[tid] = y2;
    areaS[tid] = (x2 - x1) * (y2 - y1);
  }
  for (int i = tid; i < KPAD * 7; i += 256) ((unsigned*)mask)[i] = 0u;
  if (tid < 7) keepW[tid] = 0u;
  __syncthreads();
  if (tid < KPAD && scS[tid] > SCORE_T)
    atomicOr(&keepW[tid >> 5], 1u << (tid & 31));

  // --- mask build: wave-uniform tile loop so EXEC is all-ones at the WMMA
  int lane = tid & 31, wave = tid >> 5;
  int half = lane >> 4, l = lane & 15;
  for (int t = wave; t < NTILE * NTILE; t += 8) {
    int ti = t / NTILE, tj = t % NTILE;
    int j = tj * 16 + l;
    // A 16x4 layout: lanes 0-15 hold K=0 (VGPR0) / K=1 (VGPR1); lanes 16-31 K=2/3
    v2f a;
    a.x = half ? 0.0f : areaS[ti * 16 + l];
    a.y = half ? 0.0f : 1.0f;
    // B 4x16 layout: VGPR0 lanes 0-15 = row K=0, lanes 16-31 = row K=1; VGPR1 rows 2,3
    v2f bm;
    bm.x = half ? areaS[j] : 1.0f;
    bm.y = 0.0f;
    v8f cz = {0.f, 0.f, 0.f, 0.f, 0.f, 0.f, 0.f, 0.f};
#if __has_builtin(__builtin_amdgcn_wmma_f32_16x16x4_f32)
    v8f d = __builtin_amdgcn_wmma_f32_16x16x4_f32(
        false, a, false, bm, (short)0, cz, false, false);
#else
    v8f d;
    #pragma unroll
    for (int v = 0; v < 8; v++)
      d[v] = areaS[ti * 16 + v + half * 8] + areaS[j];
#endif
    float jx1 = bx1[j], jy1 = by1[j], jx2 = bx2[j], jy2 = by2[j];
    #pragma unroll
    for (int v = 0; v < 8; v++) {
      int i = ti * 16 + v + half * 8;     // C/D layout: VGPR v -> M=v (+8 for hi half)
      float ix1 = fmaxf(bx1[i], jx1), iy1 = fmaxf(by1[i], jy1);
      float ix2 = fminf(bx2[i], jx2), iy2 = fminf(by2[i], jy2);
      float iw = fmaxf(ix2 - ix1, 0.0f), ih = fmaxf(iy2 - iy1, 0.0f);
      float inter = iw * ih;
      float uni = fmaxf(d[v] - inter, 1e-8f);   // area_i + area_j - inter
      if (inter > IOU_T * uni && i != j)
        atomicOr(&mask[i][j >> 5], 1u << (j & 31));
    }
  }
  __syncthreads();

  // --- greedy suppression, bit-parallel over 7 words; then MAX_DET cap
  if (tid == 0) {
    for (int i = 0; i < KPAD; i++) {
      if ((keepW[i >> 5] >> (i & 31)) & 1u) {
        int wi = i >> 5;
        #pragma unroll
        for (int w = 0; w < 7; w++) {
          unsigned m = mask[i][w];
          if (w < wi) m = 0u;
          else if (w == wi)
            m &= ~(((i & 31) == 31) ? 0xFFFFFFFFu : ((1u << ((i & 31) + 1)) - 1u));
          keepW[w] &= ~m;
        }
      }
    }
    int cnt = 0;
    for (int w = 0; w < 7; w++) {
      unsigned kw = keepW[w], out = 0;
      while (kw) {
        int bit = __ffs(kw) - 1;
        kw &= kw - 1;
        cnt++;
        if (cnt <= MAXDET) out |= 1u << bit;
      }
      keepW[w] = out;
    }
  }
  __syncthreads();
  if (tid < KSEL) {
    bool kp = (keepW[tid >> 5] >> (tid & 31)) & 1u;
    candScore[(size_t)bc * KSEL + tid] = kp ? scS[tid] : 0.0f;
  }
}

// ---------------------------------------------------------------- kernel 4
// One workgroup per image: top-100 of 90*200 kept scores, then emit outputs.
__global__ __launch_bounds__(256) void final_topk(
    const float* __restrict__ candScore, const float* __restrict__ candBox,
    float* __restrict__ outBoxes, float* __restrict__ outCls,
    float* __restrict__ outSc, float* __restrict__ outValid) {
  __shared__ unsigned hist[256];
  __shared__ unsigned sh_sel, sh_cum, sh_cntG, sh_cntE, sh_nv;
  __shared__ unsigned long long comp[128];
  const int M = NCLS * KSEL;
  int tid = threadIdx.x;
  int b = blockIdx.x;
  const float* base = candScore + (size_t)b * M;

  unsigned prefix = 0, maskSo = 0, remK = MAXDET;
  for (int shift = 24; shift >= 0; shift -= 8) {
    hist[tid] = 0; __syncthreads();
    for (int n = tid; n < M; n += 256) {
      unsigned key = orderKey(base[n]);
      if ((key & maskSo) == prefix) atomicAdd(&hist[(key >> shift) & 255], 1u);
    }
    __syncthreads();
    if (tid == 0) {
      unsigned cum = 0; int sel = 0;
      for (int bin = 255; bin >= 0; bin--) {
        unsigned h = hist[bin];
        if (cum + h >= remK) { sel = bin; break; }
        cum += h;
      }
      sh_sel = (unsigned)sel; sh_cum = cum;
    }
    __syncthreads();
    prefix |= (sh_sel << shift); maskSo |= (0xFFu << shift); remK -= sh_cum;
    __syncthreads();
  }
  unsigned G = MAXDET - remK;
  if (tid == 0) { sh_cntG = 0; sh_cntE = 0; sh_nv = 0; }
  if (tid < 128) comp[tid] = 0ull;
  __syncthreads();
  for (int n = tid; n < M; n += 256) {
    unsigned key = orderKey(base[n]);
    int slot = -1;
    if (key > prefix) slot = (int)atomicAdd(&sh_cntG, 1u);
    else if (key == prefix) {
      unsigned p = atomicAdd(&sh_cntE, 1u);
      if (p < remK) slot = (int)(G + p);
    }
    if (slot >= 0)
      comp[slot] = ((unsigned long long)key << 32) | (unsigned)(~(unsigned)n);
  }
  __syncthreads();
  for (int k = 2; k <= 128; k <<= 1)
    for (int j = k >> 1; j > 0; j >>= 1) {
      if (tid < 128) {
        int ixj = tid ^ j;
        if (ixj > tid) {
          unsigned long long a = comp[tid], bb = comp[ixj];
          bool up = (tid & k) != 0;
          if (up ? (a > bb) : (a < bb)) { comp[tid] = bb; comp[ixj] = a; }
        }
      }
      __syncthreads();
    }
  if (tid < MAXDET) {
    unsigned long long e = comp[tid];
    unsigned key = (unsigned)(e >> 32);
    int flat = (int)(~(unsigned)(e & 0xFFFFFFFFull));
    float sc = keyToFloat(key);
    bool valid = sc > SCORE_T;
    if (valid) atomicAdd(&sh_nv, 1u);
    const float* bp = candBox + ((size_t)b * M + flat) * 4;
    float* ob = outBoxes + ((size_t)b * MAXDET + tid) * 4;
    #pragma unroll
    for (int d4 = 0; d4 < 4; d4++) {
      float x = fminf(fmaxf(bp[d4], 0.0f), 1.0f);
      ob[d4] = valid ? x : 0.0f;
    }
    outCls[b * MAXDET + tid] = valid ? (float)(flat / KSEL) : 0.0f;
    outSc[b * MAXDET + tid]  = valid ? sc : 0.0f;
  }
  __syncthreads();
  if (tid == 0) outValid[b] = (float)sh_nv;
}

// ---------------------------------------------------------------- launcher
extern "C" void kernel_launch(void* const* d_in, const int* in_sizes, int n_in,
                              void* d_out, int out_size, void* d_ws, size_t ws_size,
                              hipStream_t stream) {
  (void)in_sizes; (void)n_in; (void)out_size; (void)ws_size;
  const float* pred    = (const float*)d_in[0];
  const float* anchors = (const float*)d_in[1];
  const float* var     = (const float*)d_in[2];

  float* ws        = (float*)d_ws;
  float* boxes     = ws;                                        // B*N*4
  float* candScore = boxes + (size_t)BATCH * NANCH * 4;         // B*C*K
  float* candBox   = candScore + (size_t)BATCH * NCLS * KSEL;   // B*C*K*4

  float* out       = (float*)d_out;
  float* outBoxes  = out;                                       // B*100*4
  float* outCls    = outBoxes + (size_t)BATCH * MAXDET * 4;     // B*100
  float* outSc     = outCls + (size_t)BATCH * MAXDET;           // B*100
  float* outValid  = outSc + (size_t)BATCH * MAXDET;            // B

  decode_boxes<<<(BATCH * NANCH + 255) / 256, 256, 0, stream>>>(pred, anchors, var, boxes);
  topk_per_class<<<BATCH * NCLS, 256, 0, stream>>>(pred, boxes, candScore, candBox);
  nms_per_class<<<BATCH * NCLS, 256, 0, stream>>>(candScore, candBox);
  final_topk<<<BATCH, 256, 0, stream>>>(candScore, candBox,
                                        outBoxes, outCls, outSc, outValid);
}